// RNNTLoss_53919019434273
// MI455X (gfx1250) — compile-verified
//
#include <hip/hip_runtime.h>
#include <stdint.h>

#define NEG_INF (-1e30f)
constexpr int V_DIM = 1024;          // vocab size (reference fixed)
constexpr int WAVES_PER_BLOCK = 8;   // 256 threads, wave32

// ---------------------------------------------------------------------------
// Async global -> LDS copy of one 4KB logits row (V_DIM floats) by one wave.
// Each lane copies 16B per issue; 8 issues cover 32*16*8 = 4096 bytes.
// gfx1250 GLOBAL_LOAD_ASYNC_TO_LDS_B128, tracked by ASYNCcnt.
// th:TH_LOAD_NT: logits (272.6 MB) are streamed exactly once and exceed the
// 192 MB L2 — non-temporal avoids thrashing L2, preserving it for the small
// blank/emit workspace that phase 2 re-reads every timestep.
// ---------------------------------------------------------------------------
__device__ __forceinline__ void async_copy_row(const float* src, uint32_t lds_off, int lane) {
  unsigned long long g = (unsigned long long)(uintptr_t)src + (unsigned long long)lane * 16ull;
  uint32_t l = lds_off + (uint32_t)lane * 16u;
#pragma unroll
  for (int k = 0; k < (V_DIM * 4) / (16 * 32); ++k) {   // 8 issues
    asm volatile("global_load_async_to_lds_b128 %0, %1, off th:TH_LOAD_NT"
                 :: "v"(l), "v"(g)
                 : "memory");
    g += 32ull * 16ull;
    l += 32u * 16u;
  }
}

// ---------------------------------------------------------------------------
// Phase 1: streaming log-softmax reduction. One wave per (b,t,u) row.
// Double-buffered async LDS pipeline; only 2 scalars survive per 4KB row.
// ---------------------------------------------------------------------------
extern "C" __global__ void __launch_bounds__(256)
rnnt_phase1(const float* __restrict__ logits,
            const int* __restrict__ targets,
            float* __restrict__ blank_out,   // (B,T,U+1)
            float* __restrict__ emit_out,    // (B,T,U)
            int B, int T, int U, int rows)
{
  __shared__ __align__(16) float tile[WAVES_PER_BLOCK][2][V_DIM];  // 64 KB
  const int lane  = threadIdx.x & 31;
  const int wv    = threadIdx.x >> 5;
  const int W     = gridDim.x * WAVES_PER_BLOCK;            // total waves
  const int gwave = blockIdx.x * WAVES_PER_BLOCK + wv;
  const int U1    = U + 1;

  const uint32_t buf_off[2] = {
    (uint32_t)(uintptr_t)&tile[wv][0][0],
    (uint32_t)(uintptr_t)&tile[wv][1][0]
  };

  if (gwave < rows)
    async_copy_row(logits + (size_t)gwave * V_DIM, buf_off[0], lane);

  for (int i = 0;; ++i) {
    const int r = gwave + i * W;
    if (r >= rows) break;
    const int rn = r + W;
    const float* cur = &tile[wv][i & 1][0];
    if (rn < rows) {
      async_copy_row(logits + (size_t)rn * V_DIM, buf_off[(i + 1) & 1], lane);
      // 8 newer loads may stay outstanding; async loads complete in order,
      // so asynccnt <= 8 guarantees row r is fully resident in LDS.
      asm volatile("s_wait_asynccnt 8" ::: "memory");
    } else {
      asm volatile("s_wait_asynccnt 0" ::: "memory");
    }

    // Two-pass logsumexp over V_DIM values: 32 floats/lane from LDS.
    float4 x[8];
#pragma unroll
    for (int k = 0; k < 8; ++k)
      x[k] = *(const float4*)&cur[(k * 32 + lane) * 4];

    float m = x[0].x;
#pragma unroll
    for (int k = 0; k < 8; ++k)
      m = fmaxf(m, fmaxf(fmaxf(x[k].x, x[k].y), fmaxf(x[k].z, x[k].w)));
#pragma unroll
    for (int off = 16; off > 0; off >>= 1)
      m = fmaxf(m, __shfl_xor(m, off, 32));

    float s = 0.f;
#pragma unroll
    for (int k = 0; k < 8; ++k)
      s += __expf(x[k].x - m) + __expf(x[k].y - m) +
           __expf(x[k].z - m) + __expf(x[k].w - m);
#pragma unroll
    for (int off = 16; off > 0; off >>= 1)
      s += __shfl_xor(s, off, 32);

    const float lse = m + __logf(s);

    if (lane == 0) {
      const int u = r % U1;
      const int t = (r / U1) % T;
      const int b = r / (U1 * T);
      blank_out[r] = cur[0] - lse;
      if (u < U) {
        const int tgt = targets[b * U + u];
        emit_out[((size_t)b * T + t) * U + u] = cur[tgt] - lse;
      }
    }
  }
}

// ---------------------------------------------------------------------------
// Phase 2: alpha recursion. One wave per batch element, 3 u-slots per lane.
// Barrier-free shfl scans (add-scan for S, logaddexp-scan for alpha).
// ---------------------------------------------------------------------------
__device__ __forceinline__ float lse2f(float a, float b) {
  const float M = fmaxf(a, b);
  return M + __logf(__expf(a - M) + __expf(b - M));
}

extern "C" __global__ void __launch_bounds__(32)
rnnt_phase2(const float* __restrict__ blank_lp,      // (B,T,U+1)
            const float* __restrict__ emit,          // (B,T,U)
            const int* __restrict__ logit_lengths,
            const int* __restrict__ target_lengths,
            float* __restrict__ ll_out,              // (B,)
            int B, int T, int U)
{
  const int b    = blockIdx.x;
  const int lane = threadIdx.x;
  const int U1   = U + 1;
  const float* blankB = blank_lp + (size_t)b * T * U1;
  const float* emitB  = emit     + (size_t)b * T * U;
  const int t_max = logit_lengths[b] - 1;
  const int u_idx = target_lengths[b];

  int u[3]; bool valU1[3], valU[3];
#pragma unroll
  for (int j = 0; j < 3; ++j) {
    u[j]     = 3 * lane + j;
    valU1[j] = u[j] < U1;
    valU[j]  = u[j] < U;
  }

  // Exclusive cumsum of emit[b,t,:] over u (S_t), via wave add-scan.
  auto compute_S = [&](int t, float S[3]) {
    float e0 = valU[0] ? emitB[(size_t)t * U + u[0]] : 0.f;
    float e1 = valU[1] ? emitB[(size_t)t * U + u[1]] : 0.f;
    float e2 = valU[2] ? emitB[(size_t)t * U + u[2]] : 0.f;
    const float l0 = e0, l1 = l0 + e1, l2 = l1 + e2;   // local inclusive
    float run = l2;
#pragma unroll
    for (int d = 1; d < 32; d <<= 1) {
      const float y = __shfl_up(run, d, 32);
      if (lane >= d) run += y;
    }
    const float pre = run - l2;                         // exclusive lane prefix
    S[0] = pre; S[1] = pre + l0; S[2] = pre + l1;       // exclusive in-lane
  };

  float alpha[3];
  {  // t = 0: alpha0[u] = S_0[u]
    float S[3]; compute_S(0, S);
    alpha[0] = S[0]; alpha[1] = S[1]; alpha[2] = S[2];
  }

  for (int t = 1; t <= t_max; ++t) {
    float S[3]; compute_S(t, S);
    float x[3];
#pragma unroll
    for (int j = 0; j < 3; ++j) {
      if (valU1[j]) {
        const float bp = blankB[(size_t)(t - 1) * U1 + u[j]];
        x[j] = alpha[j] + bp - S[j];
      } else {
        x[j] = NEG_INF;
      }
    }
    // inclusive logaddexp scan over U+1 logical elements
    const float l0 = x[0];
    const float l1 = lse2f(l0, x[1]);
    const float l2 = lse2f(l1, x[2]);
    float run = l2;
#pragma unroll
    for (int d = 1; d < 32; d <<= 1) {
      const float y = __shfl_up(run, d, 32);
      if (lane >= d) run = lse2f(y, run);
    }
    float pre = __shfl_up(run, 1, 32);                  // exclusive lane prefix
    if (lane == 0) pre = NEG_INF;
    alpha[0] = S[0] + lse2f(pre, l0);
    alpha[1] = S[1] + lse2f(pre, l1);
    alpha[2] = S[2] + lse2f(pre, l2);
  }

  const int owner = u_idx / 3, slot = u_idx % 3;
  float v = (slot == 0) ? alpha[0] : (slot == 1) ? alpha[1] : alpha[2];
  v = __shfl(v, owner, 32);
  if (lane == 0)
    ll_out[b] = v + blankB[(size_t)t_max * U1 + u_idx];
}

// ---------------------------------------------------------------------------
// Phase 3: out = -mean(ll). Overwrites d_out (deterministic per call).
// ---------------------------------------------------------------------------
extern "C" __global__ void rnnt_phase3(const float* __restrict__ ll,
                                       float* __restrict__ out, int B) {
  if (blockIdx.x == 0 && threadIdx.x == 0) {
    float s = 0.f;
    for (int i = 0; i < B; ++i) s += ll[i];
    out[0] = -s / (float)B;
  }
}

// ---------------------------------------------------------------------------
extern "C" void kernel_launch(void* const* d_in, const int* in_sizes, int n_in,
                              void* d_out, int out_size, void* d_ws, size_t ws_size,
                              hipStream_t stream)
{
  const float* logits         = (const float*)d_in[0];
  const int*   targets        = (const int*)d_in[1];
  const int*   logit_lengths  = (const int*)d_in[2];
  const int*   target_lengths = (const int*)d_in[3];

  const int B  = in_sizes[2];
  const int U  = in_sizes[1] / B;
  const int U1 = U + 1;
  const int T  = (int)((long long)in_sizes[0] / ((long long)B * U1 * V_DIM));
  const int rows = B * T * U1;

  float* blank_ws = (float*)d_ws;                       // rows floats
  float* emit_ws  = blank_ws + rows;                    // B*T*U floats
  float* ll_ws    = emit_ws + (size_t)B * T * U;        // B floats

  // 8 waves/block, each wave pipelines ~8 rows through its LDS double buffer.
  const int blocks = (rows + WAVES_PER_BLOCK * 8 - 1) / (WAVES_PER_BLOCK * 8);

  rnnt_phase1<<<blocks, WAVES_PER_BLOCK * 32, 0, stream>>>(
      logits, targets, blank_ws, emit_ws, B, T, U, rows);
  rnnt_phase2<<<B, 32, 0, stream>>>(
      blank_ws, emit_ws, logit_lengths, target_lengths, ll_ws, B, T, U);
  rnnt_phase3<<<1, 32, 0, stream>>>(ll_ws, (float*)d_out, B);
}